// AffineTransformationsToMatrix_73392401154579
// MI455X (gfx1250) — compile-verified
//
#include <hip/hip_runtime.h>

// AffineTransformationsToMatrix: per item, in = [tx,ty,tz,ax,ay,az] (f32),
// out = rowmajor 3x4 [R - I | t] with R = Rx(ax)*Ry(ay)*Rz(az).
// Memory-bound (72 B vs ~55 FLOPs per item) -> optimize for pure streaming:
// 2 items/thread => all global traffic is 16B-aligned b128, non-temporal.

typedef __attribute__((ext_vector_type(4))) float v4f;  // native clang vector (NT-builtin compatible)

__global__ __launch_bounds__(256) void
AffineToMatrix_kernel(const float* __restrict__ in,
                      float* __restrict__ out,
                      int n_items) {
    const long tid  = (long)blockIdx.x * blockDim.x + threadIdx.x;
    const long base = tid * 2;                 // first item of this thread's pair
    if (base >= n_items) return;

    const v4f* pin = (const v4f*)(in + base * 6);         // 48B = 3 x b128, 16B aligned
    v4f l0 = __builtin_nontemporal_load(pin + 0);         // t0.x t0.y t0.z a0.x
    v4f l1 = __builtin_nontemporal_load(pin + 1);         // a0.y a0.z t1.x t1.y
    v4f l2 = __builtin_nontemporal_load(pin + 2);         // t1.z a1.x a1.y a1.z

    // gfx1250 global_prefetch_b8: keep the HBM read stream ~32KB ahead.
    __builtin_prefetch(in + base * 6 + 8192, 0, 1);

    v4f* pout = (v4f*)(out + base * 12);                  // 96B = 6 x b128, 16B aligned

    // ---- item 0: t=(l0.x,l0.y,l0.z) angles=(l0.w,l1.x,l1.y) ----
    {
        float sx = __sinf(l0.w), cx = __cosf(l0.w);
        float sy = __sinf(l1.x), cy = __cosf(l1.x);
        float sz = __sinf(l1.y), cz = __cosf(l1.y);
        float sxsy = sx * sy;
        float cxsy = cx * sy;
        v4f r0 = { __builtin_fmaf(cy, cz, -1.0f),
                   -cy * sz,
                   sy,
                   l0.x };
        v4f r1 = { __builtin_fmaf(sxsy,  cz,  cx * sz),
                   __builtin_fmaf(-sxsy, sz,  cx * cz) - 1.0f,
                   -sx * cy,
                   l0.y };
        v4f r2 = { __builtin_fmaf(-cxsy, cz,  sx * sz),
                   __builtin_fmaf(cxsy,  sz,  sx * cz),
                   __builtin_fmaf(cx, cy, -1.0f),
                   l0.z };
        __builtin_nontemporal_store(r0, pout + 0);
        __builtin_nontemporal_store(r1, pout + 1);
        __builtin_nontemporal_store(r2, pout + 2);
    }

    // ---- item 1: t=(l1.z,l1.w,l2.x) angles=(l2.y,l2.z,l2.w) ----
    if (base + 1 < n_items) {
        float sx = __sinf(l2.y), cx = __cosf(l2.y);
        float sy = __sinf(l2.z), cy = __cosf(l2.z);
        float sz = __sinf(l2.w), cz = __cosf(l2.w);
        float sxsy = sx * sy;
        float cxsy = cx * sy;
        v4f r0 = { __builtin_fmaf(cy, cz, -1.0f),
                   -cy * sz,
                   sy,
                   l1.z };
        v4f r1 = { __builtin_fmaf(sxsy,  cz,  cx * sz),
                   __builtin_fmaf(-sxsy, sz,  cx * cz) - 1.0f,
                   -sx * cy,
                   l1.w };
        v4f r2 = { __builtin_fmaf(-cxsy, cz,  sx * sz),
                   __builtin_fmaf(cxsy,  sz,  sx * cz),
                   __builtin_fmaf(cx, cy, -1.0f),
                   l2.x };
        __builtin_nontemporal_store(r0, pout + 3);
        __builtin_nontemporal_store(r1, pout + 4);
        __builtin_nontemporal_store(r2, pout + 5);
    }
}

extern "C" void kernel_launch(void* const* d_in, const int* in_sizes, int n_in,
                              void* d_out, int out_size, void* d_ws, size_t ws_size,
                              hipStream_t stream) {
    const float* in  = (const float*)d_in[0];
    float*       out = (float*)d_out;

    const int  n_items = in_sizes[0] / 6;              // 4,194,304 in reference
    const long threads = ((long)n_items + 1) / 2;      // 2 items per thread
    const int  block   = 256;                          // 8 wave32 waves
    const long grid    = (threads + block - 1) / block;

    AffineToMatrix_kernel<<<dim3((unsigned)grid), dim3(block), 0, stream>>>(in, out, n_items);
}